// VectorQuantizer_3478923510114
// MI455X (gfx1250) — compile-verified
//
#include <hip/hip_runtime.h>
#include <hip/hip_bf16.h>

typedef __attribute__((ext_vector_type(2))) float v2f;
typedef __attribute__((ext_vector_type(8))) float v8f;

#define NUM_EMB 512
#define EMB_DIM 64
#define PITCH   68   // LDS row pitch (floats): 68 mod 64 = 4 -> conflict-free b64 fetches

__global__ __launch_bounds__(256) void vq_kernel(const float* __restrict__ z,
                                                 const float* __restrict__ emb,
                                                 float* __restrict__ zq,
                                                 float* __restrict__ loss,
                                                 float* __restrict__ idx_out,
                                                 int rows)
{
    __shared__ float semb[NUM_EMB * PITCH];
    __shared__ float snorm[NUM_EMB];

    const int tid = threadIdx.x;

    // Stage codebook into LDS (padded pitch) and fold per-code squared norms.
    for (int r = tid; r < NUM_EMB; r += 256) {
        float nrm = 0.f;
        #pragma unroll
        for (int i = 0; i < EMB_DIM / 4; ++i) {
            float4 v = ((const float4*)(emb + r * EMB_DIM))[i];
            ((float4*)(semb + r * PITCH))[i] = v;
            nrm += v.x*v.x + v.y*v.y + v.z*v.z + v.w*v.w;
        }
        snorm[r] = nrm;
    }
    __syncthreads();

    const int wave = tid >> 5;
    const int lane = tid & 31;
    const int mrow = lane & 15;
    const int half = lane >> 4;   // wave32 WMMA layout: half 0 -> K%4 in {0,1}, half 1 -> {2,3}
    const int koff = half * 2;

    const int row0 = (blockIdx.x * 8 + wave) * 16;
    if (row0 >= rows) return;

    // A tile: 16 z-rows x K=64 in V_WMMA_F32_16X16X4_F32 lane layout (2 VGPRs / step).
    v2f A[16];
    const float* zr = z + (size_t)(row0 + mrow) * EMB_DIM + koff;
    #pragma unroll
    for (int s = 0; s < 16; ++s)
        A[s] = *(const v2f*)(zr + 4 * s);

    float bestv[8];
    int   besti[8];
    #pragma unroll
    for (int j = 0; j < 8; ++j) { bestv[j] = __builtin_inff(); besti[j] = 0; }

    // Sweep all 512 codes, 16 at a time: 16 chained f32 WMMAs per tile (K=64).
    for (int n0 = 0; n0 < NUM_EMB; n0 += 16) {
        const int code = n0 + mrow;
        const float* br = semb + code * PITCH + koff;
        v8f acc = {};
        #pragma unroll
        for (int s = 0; s < 16; ++s) {
            v2f b = *(const v2f*)(br + 4 * s);
            acc = __builtin_amdgcn_wmma_f32_16x16x4_f32(false, A[s], false, b,
                                                        (short)0, acc, false, false);
        }
        const float en = snorm[code];
        #pragma unroll
        for (int j = 0; j < 8; ++j) {
            float metric = fmaf(-2.f, acc[j], en);   // ||e||^2 - 2 z.e  (row norm constant)
            if (metric < bestv[j]) { bestv[j] = metric; besti[j] = code; }
        }
    }

    // (min, argmin) across the 16 lanes sharing each row; first-index tie-break.
    float lsum = 0.f;
    #pragma unroll
    for (int j = 0; j < 8; ++j) {
        float v = bestv[j]; int idx = besti[j];
        #pragma unroll
        for (int off = 1; off < 16; off <<= 1) {
            float ov = __shfl_xor(v,   off, 32);
            int   oi = __shfl_xor(idx, off, 32);
            if (ov < v || (ov == v && oi < idx)) { v = ov; idx = oi; }
        }
        const int row = row0 + j + half * 8;
        // 16 lanes x 4 floats: emit z_q row from LDS codebook, fold loss partial.
        float4 e  = *(const float4*)(semb + idx * PITCH + 4 * mrow);
        float4 zv = *(const float4*)(z + (size_t)row * EMB_DIM + 4 * mrow);
        *(float4*)(zq + (size_t)row * EMB_DIM + 4 * mrow) = e;
        float dx = e.x - zv.x, dy = e.y - zv.y, dz = e.z - zv.z, dw = e.w - zv.w;
        lsum += dx*dx + dy*dy + dz*dz + dw*dw;
        if (mrow == 0) idx_out[row] = (float)idx;
    }

    // Wave-reduce loss partial, single f32 atomic per wave.
    #pragma unroll
    for (int off = 1; off < 32; off <<= 1)
        lsum += __shfl_xor(lsum, off, 32);
    if (lane == 0)
        atomicAdd(loss, lsum * (1.25f / ((float)rows * (float)EMB_DIM)));
}

__global__ void vq_zero_loss(float* loss) { *loss = 0.f; }

extern "C" void kernel_launch(void* const* d_in, const int* in_sizes, int n_in,
                              void* d_out, int out_size, void* d_ws, size_t ws_size,
                              hipStream_t stream) {
    const float* z   = (const float*)d_in[0];   // [64,4096,64] f32
    const float* emb = (const float*)d_in[1];   // [512,64] f32
    float* out  = (float*)d_out;
    const int rows = in_sizes[0] / EMB_DIM;     // 262144
    float* zq   = out;                          // rows*64 floats
    float* loss = out + (size_t)rows * EMB_DIM; // 1 float
    float* idxo = loss + 1;                     // rows values

    vq_zero_loss<<<1, 1, 0, stream>>>(loss);
    const int blocks = rows / 128;              // 8 waves/WG * 16 rows/wave
    vq_kernel<<<blocks, 256, 0, stream>>>(z, emb, zq, loss, idxo, rows);
}